// Net_24077586661451
// MI455X (gfx1250) — compile-verified
//
#include <hip/hip_runtime.h>
#include <math.h>

#define BATCH  2
#define SEQ    2048
#define NTOK   (BATCH*SEQ)     // 4096
#define DM     70
#define EDIM   140
#define NST    16
#define DTR    5
#define DBCW   (DTR + 2*NST)   // 37
#define NLAYER 12
#define INDIM  32

// padded GEMM geometry
#define KA1    72              // in_proj K (70) padded to mult of 4
#define NB1    288             // in_proj N (280) padded to mult of 16
#define KA2    140             // out_proj K, already 4-divisible
#define NB2    80              // out_proj N (70) padded to mult of 16

typedef __attribute__((ext_vector_type(2))) float v2f;
typedef __attribute__((ext_vector_type(8))) float v8f;

__device__ __forceinline__ float silu_f(float v) { return v / (1.f + __expf(-v)); }

// ---------------- one-time weight padding ----------------
// in_proj_w (L,280,70) -> (L,288,72), zero fill
__global__ void k_pad_ipw(const float* __restrict__ w, float* __restrict__ wp) {
  int i = blockIdx.x * blockDim.x + threadIdx.x;
  if (i >= NLAYER * NB1 * KA1) return;
  int c = i % KA1; int r = (i / KA1) % NB1; int l = i / (KA1 * NB1);
  wp[i] = (r < 2 * EDIM && c < DM) ? w[(l * 2 * EDIM + r) * DM + c] : 0.f;
}
// out_proj_w (L,70,140) -> (L,80,140), zero fill
__global__ void k_pad_opw(const float* __restrict__ w, float* __restrict__ wp) {
  int i = blockIdx.x * blockDim.x + threadIdx.x;
  if (i >= NLAYER * NB2 * KA2) return;
  int c = i % KA2; int r = (i / KA2) % NB2; int l = i / (KA2 * NB2);
  wp[i] = (r < DM) ? w[(l * DM + r) * KA2 + c] : 0.f;
}

// ---------------- embed: h = x @ W_in^T + b_in ----------------
__global__ void k_embed(const float* __restrict__ x, const float* __restrict__ Win,
                        const float* __restrict__ bin, float* __restrict__ h) {
  int i = blockIdx.x * blockDim.x + threadIdx.x;
  if (i >= NTOK * DM) return;
  int d = i % DM, tok = i / DM;
  float acc = bin[d];
  const float* xp = x + tok * INDIM;
  const float* wp = Win + d * INDIM;
  #pragma unroll
  for (int k = 0; k < INDIM; ++k) acc = fmaf(xp[k], wp[k], acc);
  h[i] = acc;
}

// -------- rmsnorm: one wave32 per token; writes stride-72 rows, pad=0 -----
__global__ void k_rmsnorm(const float* __restrict__ h, const float* __restrict__ w,
                          float* __restrict__ hn) {
  int wave = threadIdx.x >> 5;
  int lane = threadIdx.x & 31;
  int tok = blockIdx.x * (blockDim.x >> 5) + wave;
  const float* hp = h + tok * DM;
  float s = 0.f;
  for (int d = lane; d < DM; d += 32) { float v = hp[d]; s = fmaf(v, v, s); }
  #pragma unroll
  for (int off = 16; off > 0; off >>= 1) s += __shfl_xor(s, off, 32);
  float rs = rsqrtf(s * (1.f / (float)DM) + 1e-5f);
  float* op = hn + tok * KA1;
  for (int d = lane; d < KA1; d += 32)
    op[d] = (d < DM) ? hp[d] * rs * w[d] : 0.f;   // zero the 2 pad cols
}

// ---- fp32 WMMA GEMM on pre-padded operands: C[M x N] = A @ Bp^T (+Cadd) --
// A: rows stride lda (K padded to mult 4, pad cols zero)
// Bp: (Npad x lda) row-major, pad rows/cols zero. 4 waves/block, wave = tile.
// Branch-free inner loop: 2x global_load_b64 + 1x v_wmma per K-step.
__global__ void k_wmma_gemm(const float* __restrict__ A, const float* __restrict__ Bp,
                            float* __restrict__ C, const float* __restrict__ Cadd,
                            int N, int lda, int ksteps) {
  int lane = threadIdx.x & 31;
  int wv   = threadIdx.x >> 5;
  int half = lane >> 4;           // 0/1 -> K-pair select
  int l    = lane & 15;
  int n0 = blockIdx.x * 16;
  int m0 = (blockIdx.y * 4 + wv) * 16;
  int col = n0 + l;
  const float* ap = A  + (m0 + l) * lda + half * 2;
  const float* bp = Bp + col * lda + half * 2;
  v8f acc = {};
  #pragma unroll 4
  for (int s = 0; s < ksteps; ++s) {
    v2f a = *(const v2f*)ap;
    v2f b = *(const v2f*)bp;
    acc = __builtin_amdgcn_wmma_f32_16x16x4_f32(false, a, false, b,
                                                (short)0, acc, false, false);
    ap += 4; bp += 4;
  }
  if (col < N) {                  // single guarded store block, outside loop
    #pragma unroll
    for (int r = 0; r < 8; ++r) {
      int row = m0 + half * 8 + r;       // C/D layout: VGPR r -> row m0+half*8+r
      float v = acc[r];
      if (Cadd) v += Cadd[row * N + col];
      C[row * N + col] = v;
    }
  }
}

// ---------------- causal depthwise conv (D_CONV=4) + SiLU ----------------
__global__ void k_conv(const float* __restrict__ xz, const float* __restrict__ cw,
                       const float* __restrict__ cb, float* __restrict__ xc) {
  int i = blockIdx.x * blockDim.x + threadIdx.x;
  if (i >= NTOK * EDIM) return;
  int e = i % EDIM, tok = i / EDIM;
  int t = tok % SEQ;
  float acc = cb[e];
  const float* w = cw + e * 4;
  #pragma unroll
  for (int j = 0; j < 4; ++j) {
    int tt = t - 3 + j;
    if (tt >= 0) acc = fmaf(xz[(tok - 3 + j) * (2 * EDIM) + e], w[j], acc);
  }
  xc[i] = silu_f(acc);
}

// ---------------- x_proj: dbc = xc @ xp_w^T  (N=37, K=140) ----------------
__global__ void k_xproj(const float* __restrict__ xc, const float* __restrict__ xpw,
                        float* __restrict__ dbc) {
  int i = blockIdx.x * blockDim.x + threadIdx.x;
  if (i >= NTOK * DBCW) return;
  int r = i % DBCW, tok = i / DBCW;
  const float* xp = xc + tok * EDIM;
  const float* wp = xpw + r * EDIM;
  float acc = 0.f;
  for (int e = 0; e < EDIM; ++e) acc = fmaf(xp[e], wp[e], acc);
  dbc[i] = acc;
}

// ---------------- delta = softplus(dt @ dt_w^T + dt_b) ----------------
__global__ void k_dtproj(const float* __restrict__ dbc, const float* __restrict__ dtw,
                         const float* __restrict__ dtb, float* __restrict__ delta) {
  int i = blockIdx.x * blockDim.x + threadIdx.x;
  if (i >= NTOK * EDIM) return;
  int e = i % EDIM, tok = i / EDIM;
  float acc = dtb[e];
  const float* dp = dbc + tok * DBCW;
  const float* wp = dtw + e * DTR;
  #pragma unroll
  for (int r = 0; r < DTR; ++r) acc = fmaf(dp[r], wp[r], acc);
  delta[i] = (acc > 20.f) ? acc : log1pf(__expf(acc));
}

// ---------------- selective scan: 16 lanes per (batch, channel) ----------
__global__ void k_scan(const float* __restrict__ delta, const float* __restrict__ dbc,
                       const float* __restrict__ xc, const float* __restrict__ xz,
                       const float* __restrict__ Alog, const float* __restrict__ Dp,
                       float* __restrict__ yg) {
  int gid  = blockIdx.x * blockDim.x + threadIdx.x;
  int unit = gid >> 4;               // one 16-lane group per (b, channel)
  int n    = gid & 15;               // state index
  if (unit >= BATCH * EDIM) return;  // whole 16-lane groups retire together
  int b = unit / EDIM, e = unit % EDIM;
  float An = -__expf(Alog[e * NST + n]);   // A = -exp(A_log), hoisted
  float De = Dp[e];
  float h = 0.f;
  int base = b * SEQ;
  for (int t = 0; t < SEQ; ++t) {
    int tok = base + t;
    float dlt = delta[tok * EDIM + e];
    float xv  = xc[tok * EDIM + e];
    float Bn  = dbc[tok * DBCW + DTR + n];
    float Cn  = dbc[tok * DBCW + DTR + NST + n];
    float dA  = __expf(dlt * An);
    h = fmaf(dA, h, dlt * Bn * xv);
    float yc = Cn * h;
    yc += __shfl_xor(yc, 1, 16);
    yc += __shfl_xor(yc, 2, 16);
    yc += __shfl_xor(yc, 4, 16);
    yc += __shfl_xor(yc, 8, 16);
    if (n == 0) {
      float zv = xz[tok * 2 * EDIM + EDIM + e];
      yg[tok * EDIM + e] = (yc + De * xv) * silu_f(zv);
    }
  }
}

// ---------------- head: out = tanh(h @ W_out^T + b_out), OUT_DIM=1 -------
__global__ void k_head(const float* __restrict__ h, const float* __restrict__ Wout,
                       const float* __restrict__ bout, float* __restrict__ out) {
  int wave = threadIdx.x >> 5;
  int lane = threadIdx.x & 31;
  int tok = blockIdx.x * (blockDim.x >> 5) + wave;
  const float* hp = h + tok * DM;
  float s = 0.f;
  for (int d = lane; d < DM; d += 32) s = fmaf(hp[d], Wout[d], s);
  #pragma unroll
  for (int off = 16; off > 0; off >>= 1) s += __shfl_xor(s, off, 32);
  if (lane == 0) out[tok] = tanhf(s + bout[0]);
}

extern "C" void kernel_launch(void* const* d_in, const int* in_sizes, int n_in,
                              void* d_out, int out_size, void* d_ws, size_t ws_size,
                              hipStream_t stream) {
  (void)in_sizes; (void)n_in; (void)out_size; (void)ws_size;
  const float* x    = (const float*)d_in[0];
  const float* Win  = (const float*)d_in[1];
  const float* bin  = (const float*)d_in[2];
  const float* nw   = (const float*)d_in[3];
  const float* ipw  = (const float*)d_in[4];
  const float* cw   = (const float*)d_in[5];
  const float* cb   = (const float*)d_in[6];
  const float* xpw  = (const float*)d_in[7];
  const float* dtw  = (const float*)d_in[8];
  const float* dtb  = (const float*)d_in[9];
  const float* alog = (const float*)d_in[10];
  const float* dpp  = (const float*)d_in[11];
  const float* opw  = (const float*)d_in[12];
  const float* wout = (const float*)d_in[13];
  const float* bout = (const float*)d_in[14];
  float* out = (float*)d_out;

  float* ws = (float*)d_ws;
  float* h     = ws; ws += NTOK * DM;          // residual stream
  float* hn    = ws; ws += NTOK * KA1;         // rmsnorm output (stride 72, padded)
  float* xz    = ws; ws += NTOK * 2 * EDIM;    // in_proj output (xi | z)
  float* xc    = ws; ws += NTOK * EDIM;        // conv+silu output
  float* dbc   = ws; ws += NTOK * DBCW;        // [dt(5) | B(16) | C(16)]
  float* delta = ws; ws += NTOK * EDIM;        // softplus(dt proj)
  float* yg    = ws; ws += NTOK * EDIM;        // gated scan output
  float* ipad  = ws; ws += NLAYER * NB1 * KA1; // padded in_proj weights
  float* opad  = ws; ws += NLAYER * NB2 * KA2; // padded out_proj weights

  // one-time weight padding (zero-filled tiles -> branch-free GEMM loops)
  k_pad_ipw<<<(NLAYER * NB1 * KA1 + 255) / 256, 256, 0, stream>>>(ipw, ipad);
  k_pad_opw<<<(NLAYER * NB2 * KA2 + 255) / 256, 256, 0, stream>>>(opw, opad);

  k_embed<<<(NTOK * DM) / 256, 256, 0, stream>>>(x, Win, bin, h);

  for (int l = 0; l < NLAYER; ++l) {
    k_rmsnorm<<<NTOK / 8, 256, 0, stream>>>(h, nw + l * DM, hn);

    dim3 g1(NB1 / 16, NTOK / 64);   // 18 x 64 blocks, 4 waves/block
    k_wmma_gemm<<<g1, 128, 0, stream>>>(hn, ipad + l * NB1 * KA1, xz, nullptr,
                                        2 * EDIM, KA1, KA1 / 4);

    k_conv<<<(NTOK * EDIM) / 256, 256, 0, stream>>>(xz, cw + l * EDIM * 4,
                                                    cb + l * EDIM, xc);
    k_xproj<<<(NTOK * DBCW + 255) / 256, 256, 0, stream>>>(xc, xpw + l * DBCW * EDIM, dbc);
    k_dtproj<<<(NTOK * EDIM) / 256, 256, 0, stream>>>(dbc, dtw + l * EDIM * DTR,
                                                      dtb + l * EDIM, delta);

    k_scan<<<(BATCH * EDIM * 16 + 255) / 256, 256, 0, stream>>>(
        delta, dbc, xc, xz, alog + l * EDIM * NST, dpp + l * EDIM, yg);

    dim3 g2(NB2 / 16, NTOK / 64);   // 5 x 64 blocks, fused residual
    k_wmma_gemm<<<g2, 128, 0, stream>>>(yg, opad + l * NB2 * KA2, h, h,
                                        DM, KA2, KA2 / 4);
  }

  k_head<<<NTOK / 8, 256, 0, stream>>>(h, wout, bout, out);
}